// RNNScanTensor_12489764897123
// MI455X (gfx1250) — compile-verified
//
#include <hip/hip_runtime.h>
#include <hip/hip_bf16.h>

// Problem constants (match reference)
#define B_SZ 1024
#define T_SZ 4096
#define FD   10          // feature dim
#define NL   3           // layers
#define SD   (NL*FD)     // stacked state dim = 30
#define CHUNK 64         // timesteps per chunk
#define NC   (T_SZ/CHUNK) // 64 chunks

typedef __attribute__((ext_vector_type(2))) float    v2f;
typedef __attribute__((ext_vector_type(8))) float    v8f;
typedef __attribute__((ext_vector_type(4))) unsigned v4u;
typedef __attribute__((ext_vector_type(4))) int      v4i;
typedef __attribute__((ext_vector_type(8))) int      v8i;

// Workspace layout (floats):
//   Mp : (CHUNK+1)*SD*SD -> M^k for k=1..CHUNK (k=0 unused)
//   ES : B*NC*SD         -> chunk end-states, overwritten in place by S_c
#define MP_COUNT ((CHUNK+1)*SD*SD)

// ---------------------------------------------------------------------------
// Kernel 1: build combined 30x30 step matrix M and its powers M^1..M^CHUNK.
// ---------------------------------------------------------------------------
__global__ __launch_bounds__(1024)
void rnn_build_mpow(const float* __restrict__ W, float* __restrict__ Mp) {
  __shared__ float A[NL][FD][FD];
  __shared__ float X[NL][FD][FD];
  __shared__ float T1[FD][FD], T2[FD][FD], T3[FD][FD];
  __shared__ float Mbase[SD][SD];
  __shared__ float cur[SD][SD], nxt[SD][SD];
  const int tid = threadIdx.x;

  if (tid < NL*FD*FD) {
    int l = tid / (FD*FD), r = (tid / FD) % FD, cc = tid % FD;
    A[l][r][cc] = ((r == cc) ? 1.f : 0.f) + W[(l*FD + r)*2*FD + cc];
    X[l][r][cc] = W[(l*FD + r)*2*FD + FD + cc];
  }
  __syncthreads();
  if (tid < FD*FD) {            // T1 = W2x @ A1
    int r = tid/FD, cc = tid%FD;
    float acc = 0.f;
    for (int p = 0; p < FD; ++p) acc += X[1][r][p]*A[0][p][cc];
    T1[r][cc] = acc;
  }
  __syncthreads();
  if (tid < FD*FD) {            // T2 = W3x @ T1 ; T3 = W3x @ A2
    int r = tid/FD, cc = tid%FD;
    float a2 = 0.f, a3 = 0.f;
    for (int p = 0; p < FD; ++p) {
      a2 += X[2][r][p]*T1[p][cc];
      a3 += X[2][r][p]*A[1][p][cc];
    }
    T2[r][cc] = a2; T3[r][cc] = a3;
  }
  __syncthreads();
  if (tid < SD*SD) {            // assemble M
    int r = tid/SD, cc = tid%SD;
    float v;
    if (r < 10)       v = (cc < 10) ? A[0][r][cc] : 0.f;
    else if (r < 20) { int rr = r-10;
                       v = (cc < 10) ? T1[rr][cc] : (cc < 20 ? A[1][rr][cc-10] : 0.f); }
    else             { int rr = r-20;
                       v = (cc < 10) ? T2[rr][cc] : (cc < 20 ? T3[rr][cc-10] : A[2][rr][cc-20]); }
    Mbase[r][cc] = v; cur[r][cc] = v;
    Mp[1*SD*SD + tid] = v;
  }
  __syncthreads();
  for (int k = 2; k <= CHUNK; ++k) {
    if (tid < SD*SD) {
      int r = tid/SD, cc = tid%SD;
      float acc = 0.f;
      for (int p = 0; p < SD; ++p) acc += Mbase[r][p]*cur[p][cc];
      nxt[r][cc] = acc;
    }
    __syncthreads();
    if (tid < SD*SD) {
      int r = tid/SD, cc = tid%SD;
      cur[r][cc] = nxt[r][cc];
      Mp[k*SD*SD + tid] = nxt[r][cc];
    }
    __syncthreads();
  }
}

// ---------------------------------------------------------------------------
// Kernel 2: zero-state chunk pass. TDM prefetches the whole 16x64x10 x-slab
// into LDS (3D D#: x=feat, y=batch row @ stride T*FD, z=step @ stride FD ->
// LDS image xbuf[step][row][feat]); then 64 steps of 3 layers x 5 chained
// V_WMMA_F32_16X16X4_F32 run entirely out of LDS/registers.
// ---------------------------------------------------------------------------
__global__ __launch_bounds__(32)
void rnn_chunk_wmma(const float* __restrict__ xin, const float* __restrict__ W,
                    const float* __restrict__ bias, float* __restrict__ out,
                    float* __restrict__ ES) {
  __shared__ float xbuf[CHUNK][16][FD];   // 40 KB, filled by TDM
  __shared__ float hl[NL][16][16];        // h[layer][batch-row][feature]
  const int lane = threadIdx.x;
  const int row  = lane & 15;
  const int khi  = lane >> 4;             // 0 lanes 0-15, 1 lanes 16-31
  const int b0   = blockIdx.x * 16;
  const int c    = blockIdx.y;
  const int t0   = c * CHUNK;

  // ---- Tensor Data Mover: global (B,T,FD) slab -> LDS xbuf --------------
  {
    unsigned lds_addr = (unsigned)(unsigned long long)&xbuf[0][0][0];
    unsigned long long ga =
        (unsigned long long)(xin + ((size_t)b0*T_SZ + t0)*FD);
    v4u g0;
    g0.x = 1u;                                   // count=1 (valid descriptor)
    g0.y = lds_addr;                             // D#.lds_addr
    g0.z = (unsigned)(ga & 0xFFFFFFFFull);       // global_addr[31:0]
    g0.w = (unsigned)((ga >> 32) & 0x01FFFFFFull) | (2u << 30); // [56:32]|type=2
    v8i g1;
    g1[0] = (int)(2u << 16);                     // data_size = 4 bytes
    g1[1] = (int)((unsigned)FD << 16);           // tensor_dim0 = 10
    g1[2] = (int)(16u << 16);                    // tensor_dim1 = 16
    g1[3] = (int)((unsigned)FD << 16);           // tile_dim0 = 10
    g1[4] = (int)(16u | ((unsigned)CHUNK << 16));// tile_dim1=16, tile_dim2=64
    g1[5] = (int)(T_SZ*FD);                      // tensor_dim0_stride = 40960
    g1[6] = (int)((unsigned)FD << 16);           // tensor_dim1_stride = 10
    g1[7] = 0;
    v4i g2 = {CHUNK, 0, 0, 0};                   // tensor_dim2 = 64 (3D)
    v4i g3 = {0, 0, 0, 0};
#if defined(__clang_major__) && (__clang_major__ >= 23)
    v8i z8 = {0, 0, 0, 0, 0, 0, 0, 0};
    __builtin_amdgcn_tensor_load_to_lds(g0, g1, g2, g3, z8, 0);
#else
    __builtin_amdgcn_tensor_load_to_lds(g0, g1, g2, g3, 0);
#endif
  }

  for (int i = lane; i < NL*16*16; i += 32) ((float*)hl)[i] = 0.f;

  // Loop-invariant B fragments: B[k][n] = W_l[n][k], n = row.
  v2f bf[NL][5];
  float bb[NL];
#pragma unroll
  for (int l = 0; l < NL; ++l) {
#pragma unroll
    for (int kk = 0; kk < 5; ++kk) {
      int k0 = 4*kk + 2*khi;
      float e0 = 0.f, e1 = 0.f;
      if (row < FD) {
        e0 = W[(l*FD + row)*2*FD + k0];
        e1 = W[(l*FD + row)*2*FD + k0 + 1];
      }
      v2f v; v.x = e0; v.y = e1;
      bf[l][kk] = v;
    }
    bb[l] = (row < FD) ? bias[l*FD + row] : 0.f;
  }

  v8f hf[NL];
#pragma unroll
  for (int l = 0; l < NL; ++l)
#pragma unroll
    for (int r = 0; r < 8; ++r) hf[l][r] = 0.f;

  // Wait for the TDM fill; force xbuf's address to "escape" so the compiler
  // cannot assume the (source-invisible) TDM write never happened.
  __builtin_amdgcn_s_wait_tensorcnt((short)0);
  asm volatile("" : : "v"((void*)&xbuf[0][0][0]) : "memory");
  __syncthreads();

  for (int d = 0; d < CHUNK; ++d) {
    const int t = t0 + d;
#pragma unroll
    for (int l = 0; l < NL; ++l) {
      // A fragments from LDS: cat = [h_l (cols 0-9), inp (cols 10-19)]
      v2f af[5];
#pragma unroll
      for (int kk = 0; kk < 5; ++kk) {
        int j0 = 4*kk + 2*khi;
        float e0 = (j0 < FD) ? hl[l][row][j0]
                 : ((l == 0) ? xbuf[d][row][j0-FD] : hl[l-1][row][j0-FD]);
        int j1 = j0 + 1;
        float e1 = (j1 < FD) ? hl[l][row][j1]
                 : ((l == 0) ? xbuf[d][row][j1-FD] : hl[l-1][row][j1-FD]);
        v2f v; v.x = e0; v.y = e1;
        af[kk] = v;
      }
      v8f cacc = hf[l];
#pragma unroll
      for (int r = 0; r < 8; ++r) cacc[r] += bb[l];   // + b_l along N
#pragma unroll
      for (int kk = 0; kk < 5; ++kk) {
        cacc = __builtin_amdgcn_wmma_f32_16x16x4_f32(
            false, af[kk], false, bf[l][kk], (short)0, cacc, false, false);
      }
      hf[l] = cacc;
#pragma unroll
      for (int r = 0; r < 8; ++r)            // D -> LDS (batch = r+8*khi, feat = row)
        hl[l][r + 8*khi][row] = cacc[r];
      __syncthreads();
    }
    // emit layer-3 zero-state output
    if (row < FD) {
#pragma unroll
      for (int r = 0; r < 8; ++r)
        out[((size_t)t*B_SZ + (b0 + r + 8*khi))*FD + row] = hf[NL-1][r];
    }
    __syncthreads();
  }
  // chunk-end zero-state (30 comps per batch row) for the boundary scan
  for (int i = lane; i < 16*SD; i += 32) {
    int lb = i / SD, comp = i % SD;
    ES[((size_t)(b0 + lb)*NC + c)*SD + comp] = hl[comp/FD][lb][comp%FD];
  }
}

// ---------------------------------------------------------------------------
// Kernel 3: sequential boundary scan over chunks (per batch element).
//   S_0 = initial[b];  ES[b][c] <- S_c (in place);  S_{c+1} = M^L S_c + E_c
// ---------------------------------------------------------------------------
__global__ __launch_bounds__(256)
void rnn_boundary(const float* __restrict__ initial, const float* __restrict__ Mp,
                  float* __restrict__ ES) {
  __shared__ float ML[SD][SD];
  const int tid = threadIdx.x;
  for (int i = tid; i < SD*SD; i += 256) ((float*)ML)[i] = Mp[CHUNK*SD*SD + i];
  __syncthreads();
  const int b = blockIdx.x*256 + tid;
  float s[SD];
#pragma unroll
  for (int i = 0; i < SD; ++i) s[i] = initial[(size_t)b*SD + i];
  for (int c = 0; c < NC; ++c) {
    const size_t base = ((size_t)b*NC + c)*SD;
    float e[SD];
#pragma unroll
    for (int i = 0; i < SD; ++i) e[i] = ES[base + i];   // zero-state end
#pragma unroll
    for (int i = 0; i < SD; ++i) ES[base + i] = s[i];   // overwrite with S_c
    float ns[SD];
    for (int i = 0; i < SD; ++i) {
      float acc = e[i];
#pragma unroll
      for (int j = 0; j < SD; ++j) acc += ML[i][j]*s[j];
      ns[i] = acc;
    }
#pragma unroll
    for (int i = 0; i < SD; ++i) s[i] = ns[i];
  }
}

// ---------------------------------------------------------------------------
// Kernel 4: correction  y[b, c*L+d] += (M^{d+1} S_c)[20:30].
// blockIdx.y = t, so the 10x30 row-slice of M^{d+1} is block-uniform -> LDS.
// Fully coalesced RMW on y (10 consecutive floats per thread).
// ---------------------------------------------------------------------------
__global__ __launch_bounds__(256)
void rnn_correct(const float* __restrict__ Mp, const float* __restrict__ ES,
                 float* __restrict__ out) {
  __shared__ float R[FD][SD];
  const int tid = threadIdx.x;
  const int t = blockIdx.y;
  const int c = t / CHUNK, d = t % CHUNK;
  const int k = d + 1;
  for (int i = tid; i < FD*SD; i += 256)
    ((float*)R)[i] = Mp[k*SD*SD + 20*SD + i];          // rows 20..29 of M^k
  __syncthreads();
  const int b = blockIdx.x*256 + tid;
  float s[SD];
  const size_t sb = ((size_t)b*NC + c)*SD;
#pragma unroll
  for (int i = 0; i < SD; ++i) s[i] = ES[sb + i];
  const size_t ob = ((size_t)t*B_SZ + b)*FD;
#pragma unroll
  for (int f = 0; f < FD; ++f) {
    float acc = out[ob + f];
    for (int j = 0; j < SD; ++j) acc += R[f][j]*s[j];
    out[ob + f] = acc;
  }
}

// ---------------------------------------------------------------------------
extern "C" void kernel_launch(void* const* d_in, const int* in_sizes, int n_in,
                              void* d_out, int out_size, void* d_ws, size_t ws_size,
                              hipStream_t stream) {
  (void)in_sizes; (void)n_in; (void)out_size; (void)ws_size;
  const float* initial = (const float*)d_in[0];   // (B, 3, 10)
  const float* xin     = (const float*)d_in[1];   // (B, T, 10)
  const float* W       = (const float*)d_in[2];   // (3, 10, 20)
  const float* bias    = (const float*)d_in[3];   // (3, 10)
  float* out = (float*)d_out;                     // (T, B, 10)
  float* ws  = (float*)d_ws;
  float* Mp  = ws;                                // 65*900 floats
  float* ES  = ws + MP_COUNT;                     // B*NC*30 floats (~7.9 MB)

  rnn_build_mpow<<<1, 1024, 0, stream>>>(W, Mp);
  rnn_chunk_wmma<<<dim3(B_SZ/16, NC), 32, 0, stream>>>(xin, W, bias, out, ES);
  rnn_boundary<<<dim3(B_SZ/256), 256, 0, stream>>>(initial, Mp, ES);
  rnn_correct<<<dim3(B_SZ/256, T_SZ), 256, 0, stream>>>(Mp, ES, out);
}